// TestModule_11716670783755
// MI455X (gfx1250) — compile-verified
//
#include <hip/hip_runtime.h>
#include <hip/hip_bf16.h>
#include <math.h>

#define NNODE 20000
#define NEDGE 200000
#define FIN   64
#define HDIM  32
#define CDIM  10
#define CPAD  16

typedef __attribute__((ext_vector_type(2))) float v2f;
typedef __attribute__((ext_vector_type(8))) float v8f;

// ---------------------------------------------------------------------------
// Weight prep: combine per-dst-type root weights/biases, pad layer-2 weights
// ---------------------------------------------------------------------------
__global__ void prep_weights(const float* __restrict__ W1_root, const float* __restrict__ b1,
                             const float* __restrict__ W2_rel,  const float* __restrict__ b2,
                             const float* __restrict__ W2_root,
                             float* __restrict__ Wr1c, float* __restrict__ b1c,
                             float* __restrict__ W2relp, float* __restrict__ W2rootc,
                             float* __restrict__ b2c) {
    const int ra[3] = {1, 0, 2};   // relations with dst type t
    const int rb[3] = {3, 5, 4};
    int tid = threadIdx.x;
    for (int i = tid; i < 3 * FIN * HDIM; i += blockDim.x) {
        int t = i / (FIN * HDIM), j = i % (FIN * HDIM);
        Wr1c[i] = W1_root[ra[t] * FIN * HDIM + j] + W1_root[rb[t] * FIN * HDIM + j];
    }
    for (int i = tid; i < 3 * HDIM; i += blockDim.x) {
        int t = i / HDIM, j = i % HDIM;
        b1c[i] = b1[ra[t] * HDIM + j] + b1[rb[t] * HDIM + j];
    }
    const int rr[2] = {1, 3};      // only relations feeding node type 0
    for (int i = tid; i < 2 * HDIM * CPAD; i += blockDim.x) {
        int q = i / (HDIM * CPAD), j = i % (HDIM * CPAD);
        int row = j / CPAD, col = j % CPAD;
        W2relp[i] = (col < CDIM) ? W2_rel[rr[q] * HDIM * CDIM + row * CDIM + col] : 0.f;
    }
    for (int i = tid; i < HDIM * CPAD; i += blockDim.x) {
        int row = i / CPAD, col = i % CPAD;
        W2rootc[i] = (col < CDIM)
            ? (W2_root[1 * HDIM * CDIM + row * CDIM + col] +
               W2_root[3 * HDIM * CDIM + row * CDIM + col]) : 0.f;
    }
    for (int i = tid; i < CPAD; i += blockDim.x)
        b2c[i] = (i < CDIM) ? (b2[1 * CDIM + i] + b2[3 * CDIM + i]) : 0.f;
}

// ---------------------------------------------------------------------------
// GEMM  Out[nrows x 32] = X[nrows x 64] @ W[64 x 32] (+bias), fp32 WMMA
// One wave computes a 16x32 output tile: two 16x16 WMMA accumulators,
// 16 k-steps of V_WMMA_F32_16X16X4_F32.
// ---------------------------------------------------------------------------
__global__ __launch_bounds__(256) void gemm_k64_n32(const float* __restrict__ X,
                                                    const float* __restrict__ W,
                                                    const float* __restrict__ bias,
                                                    float* __restrict__ Out, int nrows) {
    const int lane = threadIdx.x & 31;
    const int wave = threadIdx.x >> 5;
    const int hi   = lane >> 4;           // 0: K pair {0,1}, 1: K pair {2,3}
    const int lo   = lane & 15;

    v2f b0[16], b1f[16];                  // B fragments for n-tiles 0 and 1
    #pragma unroll
    for (int kk = 0; kk < 16; ++kk) {
        int krow = 4 * kk + 2 * hi;
        b0[kk].x  = W[krow * HDIM + lo];
        b0[kk].y  = W[(krow + 1) * HDIM + lo];
        b1f[kk].x = W[krow * HDIM + 16 + lo];
        b1f[kk].y = W[(krow + 1) * HDIM + 16 + lo];
    }
    float bv0 = 0.f, bv1 = 0.f;
    if (bias) { bv0 = bias[lo]; bv1 = bias[16 + lo]; }

    const int ntiles = nrows >> 4;
    for (int tile = blockIdx.x * 8 + wave; tile < ntiles; tile += gridDim.x * 8) {
        const int mbase = tile << 4;
        v8f c0 = {}; v8f c1 = {};
        const float* xrow = X + (size_t)(mbase + lo) * FIN + 2 * hi;
        #pragma unroll
        for (int kk = 0; kk < 16; ++kk) {
            v2f a;
            a.x = xrow[4 * kk];
            a.y = xrow[4 * kk + 1];
            c0 = __builtin_amdgcn_wmma_f32_16x16x4_f32(false, a, false, b0[kk],
                                                       (short)0, c0, false, false);
            c1 = __builtin_amdgcn_wmma_f32_16x16x4_f32(false, a, false, b1f[kk],
                                                       (short)0, c1, false, false);
        }
        float* orow = Out + (size_t)mbase * HDIM;
        #pragma unroll
        for (int j = 0; j < 8; ++j) {
            int m = j + 8 * hi;
            orow[m * HDIM + lo]      = c0[j] + bv0;
            orow[m * HDIM + 16 + lo] = c1[j] + bv1;
        }
    }
}

// ---------------------------------------------------------------------------
// GEMM  Out[nrows x 16] = X[nrows x 32] @ W[32 x 16] (+bias), fp32 WMMA
// ---------------------------------------------------------------------------
__global__ __launch_bounds__(256) void gemm_k32_n16(const float* __restrict__ X,
                                                    const float* __restrict__ W,
                                                    const float* __restrict__ bias,
                                                    float* __restrict__ Out, int nrows) {
    const int lane = threadIdx.x & 31;
    const int wave = threadIdx.x >> 5;
    const int hi   = lane >> 4;
    const int lo   = lane & 15;

    v2f bf[8];
    #pragma unroll
    for (int kk = 0; kk < 8; ++kk) {
        int krow = 4 * kk + 2 * hi;
        bf[kk].x = W[krow * CPAD + lo];
        bf[kk].y = W[(krow + 1) * CPAD + lo];
    }
    float bv = bias ? bias[lo] : 0.f;

    const int ntiles = nrows >> 4;
    for (int tile = blockIdx.x * 8 + wave; tile < ntiles; tile += gridDim.x * 8) {
        const int mbase = tile << 4;
        v8f c = {};
        const float* xrow = X + (size_t)(mbase + lo) * HDIM + 2 * hi;
        #pragma unroll
        for (int kk = 0; kk < 8; ++kk) {
            v2f a;
            a.x = xrow[4 * kk];
            a.y = xrow[4 * kk + 1];
            c = __builtin_amdgcn_wmma_f32_16x16x4_f32(false, a, false, bf[kk],
                                                      (short)0, c, false, false);
        }
        float* orow = Out + (size_t)mbase * CPAD;
        #pragma unroll
        for (int j = 0; j < 8; ++j)
            orow[(j + 8 * hi) * CPAD + lo] = c[j] + bv;
    }
}

// ---------------------------------------------------------------------------
// Edge scatter-add: 8 threads/edge, float4 gather, 4 f32 atomics each (H=32)
// ---------------------------------------------------------------------------
__global__ __launch_bounds__(256) void scatter_h32(const int* __restrict__ ei,
                                                   const float* __restrict__ T,
                                                   float* __restrict__ Hdst) {
    int idx = blockIdx.x * blockDim.x + threadIdx.x;
    if (idx >= NEDGE * 8) return;
    int e  = idx >> 3;
    int cg = (idx & 7) * 4;
    int src = ei[e];
    int dst = ei[NEDGE + e];
    const float4 v = *(const float4*)(T + (size_t)src * HDIM + cg);
    float* p = Hdst + (size_t)dst * HDIM + cg;
    atomicAdd(p + 0, v.x); atomicAdd(p + 1, v.y);
    atomicAdd(p + 2, v.z); atomicAdd(p + 3, v.w);
}

// 4 threads/edge for padded-16 logits rows
__global__ __launch_bounds__(256) void scatter_l16(const int* __restrict__ ei,
                                                   const float* __restrict__ T,
                                                   float* __restrict__ L) {
    int idx = blockIdx.x * blockDim.x + threadIdx.x;
    if (idx >= NEDGE * 4) return;
    int e  = idx >> 2;
    int cg = (idx & 3) * 4;
    int src = ei[e];
    int dst = ei[NEDGE + e];
    const float4 v = *(const float4*)(T + (size_t)src * CPAD + cg);
    float* p = L + (size_t)dst * CPAD + cg;
    atomicAdd(p + 0, v.x); atomicAdd(p + 1, v.y);
    atomicAdd(p + 2, v.z); atomicAdd(p + 3, v.w);
}

__global__ __launch_bounds__(256) void relu_vec4(float* __restrict__ h, int n4) {
    int i = blockIdx.x * blockDim.x + threadIdx.x;
    if (i >= n4) return;
    float4 v = ((float4*)h)[i];
    v.x = fmaxf(v.x, 0.f); v.y = fmaxf(v.y, 0.f);
    v.z = fmaxf(v.z, 0.f); v.w = fmaxf(v.w, 0.f);
    ((float4*)h)[i] = v;
}

// ---------------------------------------------------------------------------
// NLL loss over type-0 nodes: per-block partial sums (deterministic reduce)
// ---------------------------------------------------------------------------
__global__ __launch_bounds__(256) void loss_partial(const float* __restrict__ logits,
                                                    const int* __restrict__ y,
                                                    float* __restrict__ partials, int n) {
    __shared__ float s[256];
    int i = blockIdx.x * 256 + threadIdx.x;
    float v = 0.f;
    if (i < n) {
        const float* l = logits + (size_t)i * CPAD;
        float m = l[0];
        #pragma unroll
        for (int c = 1; c < CDIM; ++c) m = fmaxf(m, l[c]);
        float sum = 0.f;
        #pragma unroll
        for (int c = 0; c < CDIM; ++c) sum += expf(l[c] - m);
        v = m + logf(sum) - l[y[i]];
    }
    s[threadIdx.x] = v;
    __syncthreads();
    for (int off = 128; off > 0; off >>= 1) {
        if (threadIdx.x < off) s[threadIdx.x] += s[threadIdx.x + off];
        __syncthreads();
    }
    if (threadIdx.x == 0) partials[blockIdx.x] = s[0];
}

__global__ void loss_final(const float* __restrict__ partials, int nblocks,
                           float* __restrict__ out) {
    float v = 0.f;
    for (int i = threadIdx.x; i < nblocks; i += 32) v += partials[i];
    for (int off = 16; off > 0; off >>= 1) v += __shfl_down(v, off, 32);
    if (threadIdx.x == 0) out[0] = v / (float)NNODE;
}

// ---------------------------------------------------------------------------
extern "C" void kernel_launch(void* const* d_in, const int* in_sizes, int n_in,
                              void* d_out, int out_size, void* d_ws, size_t ws_size,
                              hipStream_t stream) {
    const float* x0      = (const float*)d_in[0];
    const float* x1      = (const float*)d_in[1];
    const float* x2      = (const float*)d_in[2];
    const float* W1_rel  = (const float*)d_in[3];
    const float* b1      = (const float*)d_in[4];
    const float* W1_root = (const float*)d_in[5];
    const float* W2_rel  = (const float*)d_in[6];
    const float* b2      = (const float*)d_in[7];
    const float* W2_root = (const float*)d_in[8];
    const int*   y       = (const int*)d_in[9];
    const int*   ei[6]   = {(const int*)d_in[10], (const int*)d_in[11], (const int*)d_in[12],
                            (const int*)d_in[13], (const int*)d_in[14], (const int*)d_in[15]};
    const float* xs[3]   = {x0, x1, x2};

    // workspace layout (floats)
    float* ws = (float*)d_ws;
    size_t off = 0;
    float* T1      = ws + off; off += (size_t)6 * NNODE * HDIM;   // per-rel transformed sources
    float* h       = ws + off; off += (size_t)3 * NNODE * HDIM;   // hidden per node type
    float* T2      = ws + off; off += (size_t)2 * NNODE * CPAD;   // layer-2 transformed (r=1,3)
    float* logits  = ws + off; off += (size_t)NNODE * CPAD;
    float* Wr1c    = ws + off; off += 3 * FIN * HDIM;
    float* b1c     = ws + off; off += 3 * HDIM;
    float* W2relp  = ws + off; off += 2 * HDIM * CPAD;
    float* W2rootc = ws + off; off += HDIM * CPAD;
    float* b2c     = ws + off; off += CPAD;
    float* partials= ws + off; off += 256;

    prep_weights<<<1, 256, 0, stream>>>(W1_root, b1, W2_rel, b2, W2_root,
                                        Wr1c, b1c, W2relp, W2rootc, b2c);

    const int srct[6] = {0, 1, 0, 2, 1, 2};
    const int dstt[6] = {1, 0, 2, 0, 2, 1};
    const int gemm_blocks = (NNODE / 16 + 7) / 8;     // 157, grid-stride over 1250 tiles

    // Layer 1: transform sources per relation (x @ W1_rel[r])
    for (int r = 0; r < 6; ++r)
        gemm_k64_n32<<<gemm_blocks, 256, 0, stream>>>(
            xs[srct[r]], W1_rel + (size_t)r * FIN * HDIM, nullptr,
            T1 + (size_t)r * NNODE * HDIM, NNODE);

    // Layer 1: root term initializes h[t] = x[t] @ (sum W1_root) + sum b1
    for (int t = 0; t < 3; ++t)
        gemm_k64_n32<<<gemm_blocks, 256, 0, stream>>>(
            xs[t], Wr1c + (size_t)t * FIN * HDIM, b1c + t * HDIM,
            h + (size_t)t * NNODE * HDIM, NNODE);

    // Layer 1: edge scatter-add of transformed rows
    const int sc1_blocks = (NEDGE * 8 + 255) / 256;
    for (int r = 0; r < 6; ++r)
        scatter_h32<<<sc1_blocks, 256, 0, stream>>>(
            ei[r], T1 + (size_t)r * NNODE * HDIM, h + (size_t)dstt[r] * NNODE * HDIM);

    relu_vec4<<<(3 * NNODE * HDIM / 4 + 255) / 256, 256, 0, stream>>>(h, 3 * NNODE * HDIM / 4);

    // Layer 2: only relations (1->0) [r=1] and (2->0) [r=3] reach the loss
    gemm_k32_n16<<<gemm_blocks, 256, 0, stream>>>(
        h + (size_t)1 * NNODE * HDIM, W2relp, nullptr, T2, NNODE);
    gemm_k32_n16<<<gemm_blocks, 256, 0, stream>>>(
        h + (size_t)2 * NNODE * HDIM, W2relp + HDIM * CPAD, nullptr,
        T2 + (size_t)NNODE * CPAD, NNODE);
    gemm_k32_n16<<<gemm_blocks, 256, 0, stream>>>(
        h, W2rootc, b2c, logits, NNODE);

    const int sc2_blocks = (NEDGE * 4 + 255) / 256;
    scatter_l16<<<sc2_blocks, 256, 0, stream>>>(ei[1], T2, logits);
    scatter_l16<<<sc2_blocks, 256, 0, stream>>>(ei[3], T2 + (size_t)NNODE * CPAD, logits);

    const int nb = (NNODE + 255) / 256;               // 79
    loss_partial<<<nb, 256, 0, stream>>>(logits, y, partials, NNODE);
    loss_final<<<1, 32, 0, stream>>>(partials, nb, (float*)d_out);
}